// NGCF_22222160790059
// MI455X (gfx1250) — compile-verified
//
#include <hip/hip_runtime.h>

typedef __attribute__((ext_vector_type(2))) float v2f;
typedef __attribute__((ext_vector_type(8))) float v8f;

#define EMB 64

// ---------------------------------------------------------------- zero
__global__ void ngcf_zero(float4* __restrict__ p, int n4) {
    int i = blockIdx.x * blockDim.x + threadIdx.x;
    if (i < n4) p[i] = make_float4(0.f, 0.f, 0.f, 0.f);
}

// ---------------------------------------------------------------- SpMM
// side[row] += val * src[col]   (COO gather-multiply-scatter, one wave/edge)
// src rows are 64 floats at stride `srcStride`.
__global__ void ngcf_spmm(const int* __restrict__ rows,
                          const int* __restrict__ cols,
                          const float* __restrict__ vals,
                          const float* __restrict__ src, int srcStride,
                          float* __restrict__ side, int nnz) {
    int wid  = (blockIdx.x * blockDim.x + threadIdx.x) >> 5;
    int lane = threadIdx.x & 31;
    if (wid >= nnz) return;
    int   r = rows[wid];
    int   c = cols[wid];
    float v = vals[wid];
    const float2* s2 = (const float2*)(src + (size_t)c * srcStride);
    float2 x = s2[lane];
    float* dst = side + (size_t)r * EMB + 2 * lane;
    unsafeAtomicAdd(dst,     v * x.x);
    unsafeAtomicAdd(dst + 1, v * x.y);
}

// ---------------------------------------------------------------- dense
// Per node i:  e = leaky( (x+s)@Gw^T + (x*s)@Bw^T + (gb+bb), 0.2 )
//              norm = e / max(||e||2, 1e-12)
// LAYER==0: x from emb[i*64]; writes out[i,0:64]=x, out[i,64:128]=norm,
//           out[i,128:192]=e (raw ego, consumed by layer-1 SpMM).
// LAYER==1: x from out[i,128:192]; overwrites out[i,128:192]=norm.
template <int LAYER>
__global__ __launch_bounds__(128)
void ngcf_dense(const float* __restrict__ emb,
                const float* __restrict__ side,
                const float* __restrict__ gw, const float* __restrict__ gb,
                const float* __restrict__ bw, const float* __restrict__ bb,
                float* __restrict__ out, int ntiles) {
    // LDS: W^T with K-pairs packed: s[p*128 + 2*j + t] = W[j*64 + 2p + t]
    __shared__ float sG[32 * 128];
    __shared__ float sB[32 * 128];
    for (int idx = threadIdx.x; idx < 4096; idx += 128) {
        int j = idx >> 6, kk = idx & 63;
        int o = (kk >> 1) * 128 + (j << 1) + (kk & 1);
        sG[o] = gw[idx];
        sB[o] = bw[idx];
    }
    __syncthreads();

    int tile = blockIdx.x * 4 + (threadIdx.x >> 5);
    if (tile >= ntiles) return;               // wave-uniform exit; EXEC stays full

    const int lane = threadIdx.x & 31;
    const int half = lane >> 4;               // K sub-pair / M half selector
    const int mr   = lane & 15;               // A-frag row, B/C-frag column
    const int tileBase = tile * 16;
    const int nodeA = tileBase + mr;

    const float* xrow = (LAYER == 0) ? emb + (size_t)nodeA * EMB
                                     : out + (size_t)nodeA * 192 + 128;
    const float* srow = side + (size_t)nodeA * EMB;

    v8f c[4] = {};                            // 4 N-tiles of 16x16 f32 accum

    for (int k0 = 0; k0 < 64; k0 += 4) {
        int k = k0 + half * 2;
        float2 x2 = *(const float2*)(xrow + k);
        float2 s2 = *(const float2*)(srow + k);
        if (LAYER == 0)                        // fold emb copy into out[:,0:64]
            *(float2*)(out + (size_t)nodeA * 192 + k) = x2;
        v2f asum = {x2.x + s2.x, x2.y + s2.y};
        v2f abi  = {x2.x * s2.x, x2.y * s2.y};
        const int prow = ((k0 >> 1) + half) * 128;
#pragma unroll
        for (int jt = 0; jt < 4; jt++) {
            int boff = prow + ((jt * 16 + mr) << 1);
            v2f bg = *(const v2f*)&sG[boff];
            v2f bi = *(const v2f*)&sB[boff];
            c[jt] = __builtin_amdgcn_wmma_f32_16x16x4_f32(
                false, asum, false, bg, (short)0, c[jt], false, false);
            c[jt] = __builtin_amdgcn_wmma_f32_16x16x4_f32(
                false, abi, false, bi, (short)0, c[jt], false, false);
        }
    }

    // bias (gb+bb), leaky-relu, per-row sum of squares
    float bias[4];
#pragma unroll
    for (int jt = 0; jt < 4; jt++)
        bias[jt] = gb[jt * 16 + mr] + bb[jt * 16 + mr];

    float ssq[8];
#pragma unroll
    for (int r = 0; r < 8; r++) {
        float s = 0.f;
#pragma unroll
        for (int jt = 0; jt < 4; jt++) {
            float v = c[jt][r] + bias[jt];
            v = (v > 0.f) ? v : 0.2f * v;
            c[jt][r] = v;
            s += v * v;
        }
        ssq[r] = s;
    }
    // reduce across the 16 lanes of each half (rows m = r + half*8)
#pragma unroll
    for (int r = 0; r < 8; r++) {
        float s = ssq[r];
        s += __shfl_xor(s, 1, 32);
        s += __shfl_xor(s, 2, 32);
        s += __shfl_xor(s, 4, 32);
        s += __shfl_xor(s, 8, 32);
        ssq[r] = s;
    }

#pragma unroll
    for (int r = 0; r < 8; r++) {
        int m = r + half * 8;
        size_t base = (size_t)(tileBase + m) * 192;
        float sc = 1.0f / fmaxf(sqrtf(ssq[r]), 1e-12f);
#pragma unroll
        for (int jt = 0; jt < 4; jt++) {
            int j = jt * 16 + mr;
            float v = c[jt][r];
            if (LAYER == 0) {
                out[base + 64 + j]  = v * sc;  // norm1
                out[base + 128 + j] = v;       // raw ego for layer 1
            } else {
                out[base + 128 + j] = v * sc;  // norm2
            }
        }
    }
}

// ---------------------------------------------------------------- launch
extern "C" void kernel_launch(void* const* d_in, const int* in_sizes, int n_in,
                              void* d_out, int out_size, void* d_ws, size_t ws_size,
                              hipStream_t stream) {
    (void)n_in; (void)out_size; (void)ws_size;
    const float* emb   = (const float*)d_in[0];
    const float* gc_w0 = (const float*)d_in[1];
    const float* gc_b0 = (const float*)d_in[2];
    const float* bi_w0 = (const float*)d_in[3];
    const float* bi_b0 = (const float*)d_in[4];
    const float* gc_w1 = (const float*)d_in[5];
    const float* gc_b1 = (const float*)d_in[6];
    const float* bi_w1 = (const float*)d_in[7];
    const float* bi_b1 = (const float*)d_in[8];
    const int*   rows  = (const int*)d_in[9];
    const int*   cols  = (const int*)d_in[10];
    const float* vals  = (const float*)d_in[11];

    float* out  = (float*)d_out;
    float* side = (float*)d_ws;          // N*64 f32 = 76.8 MB scratch

    const int N      = in_sizes[0] / EMB;    // 300000
    const int NNZ    = in_sizes[9];          // 9.6M
    const int ntiles = N / 16;               // 18750 (N divisible by 16)

    const int n4 = N * EMB / 4;
    dim3 zb((n4 + 255) / 256), zt(256);
    dim3 sb((NNZ + 7) / 8), st(256);         // 8 waves/block, 1 edge/wave
    dim3 db((ntiles + 3) / 4), dt(128);      // 4 waves/block, 16 nodes/wave

    // ---- layer 0
    ngcf_zero<<<zb, zt, 0, stream>>>((float4*)side, n4);
    ngcf_spmm<<<sb, st, 0, stream>>>(rows, cols, vals, emb, EMB, side, NNZ);
    ngcf_dense<0><<<db, dt, 0, stream>>>(emb, side, gc_w0, gc_b0, bi_w0, bi_b0,
                                         out, ntiles);
    // ---- layer 1 (ego lives in out[:,128:192], stride 192)
    ngcf_zero<<<zb, zt, 0, stream>>>((float4*)side, n4);
    ngcf_spmm<<<sb, st, 0, stream>>>(rows, cols, vals, out + 128, 192, side, NNZ);
    ngcf_dense<1><<<db, dt, 0, stream>>>(emb, side, gc_w1, gc_b1, bi_w1, bi_b1,
                                         out, ntiles);
}